// DigitCap_12412455486125
// MI455X (gfx1250) — compile-verified
//
#include <hip/hip_runtime.h>
#include <hip/hip_bf16.h>

// DigitCaps layer, attention branch eliminated (softmax over size-1 axis == 1):
//   S[b,m,d] = sum_{n,p} (1 + Bp[m,n]) * W[m,n,d,p] * u[b,n,p];  out = squash(S)
// Implemented as 10 independent 16x16x9216 FP32 GEMMs via v_wmma_f32_16x16x4_f32.

typedef __attribute__((ext_vector_type(2))) float v2f;
typedef __attribute__((ext_vector_type(8))) float v8f;

#define NWAVES 8
#define NCAPS  1152
#define DP     8
#define BATCH  16
#define DD     16
#define MCAPS  10
#define KTOT   (NCAPS * DP)   // 9216

__global__ __launch_bounds__(256)
void digitcaps_wmma_kernel(const float* __restrict__ u,   // [16,1152,8]
                           const float* __restrict__ W,   // [10,1152,16,8]
                           const float* __restrict__ Bp,  // [10,1,1152]
                           float* __restrict__ out)       // [16,10,16]
{
    __shared__ float red[NWAVES][BATCH * DD];  // per-wave partial C tiles
    __shared__ float sq[BATCH * DD];           // squared elements for squash norm

    const int m    = blockIdx.x;               // digit capsule 0..9
    const int tid  = threadIdx.x;
    const int wave = tid >> 5;
    const int lane = tid & 31;
    const int half = lane >> 4;                // 0: lanes 0-15, 1: lanes 16-31
    const int l16  = lane & 15;

    const int nPerWave = NCAPS / NWAVES;       // 144 primary caps per wave
    const int n0 = wave * nPerWave;

    const float* __restrict__ Wm  = W  + (size_t)m * NCAPS * DD * DP;
    const float* __restrict__ Bpm = Bp + (size_t)m * NCAPS;
    // A-matrix row for this lane: M = batch index = l16, contiguous K = n*8+p
    const float* __restrict__ uRow = u + (size_t)l16 * KTOT;

    v8f acc = {0.f, 0.f, 0.f, 0.f, 0.f, 0.f, 0.f, 0.f};

    for (int n = n0; n < n0 + nPerWave; ++n) {
        const float s = 1.0f + Bpm[n];                 // CB = 1 + B_prior (C == 1)
        // B fragment base: W[m][n][d=l16][p], K-offset 2*half per ISA A/B layout
        const float* __restrict__ wn = Wm + (size_t)n * (DD * DP) + l16 * DP + 2 * half;
        const float* __restrict__ an = uRow + (size_t)n * DP + 2 * half;
        __builtin_prefetch(wn + 4 * DD * DP, 0, 1);    // global_prefetch_b8 on W stream
#pragma unroll
        for (int pb = 0; pb < DP; pb += 4) {           // two K=4 steps per primary cap
            v2f a;
            a.x = an[pb];                              // A[M=l16][K0 + 2*half + 0]
            a.y = an[pb + 1];                          // A[M=l16][K0 + 2*half + 1]
            v2f b;
            b.x = s * wn[pb];                          // B[K0 + 2*half + 0][N=l16]
            b.y = s * wn[pb + 1];                      // B[K0 + 2*half + 1][N=l16]
            acc = __builtin_amdgcn_wmma_f32_16x16x4_f32(
                false, a, false, b, (short)0, acc, false, false);
        }
    }

    // Spill per-wave 16x16 accumulator: C layout VGPR j -> M = j + 8*half, N = l16
#pragma unroll
    for (int j = 0; j < 8; ++j) {
        const int brow = j + 8 * half;                 // batch index
        red[wave][brow * DD + l16] = acc[j];
    }
    __syncthreads();

    // Cross-wave reduction: thread t owns (b = t/16, d = t%16)
    float sv = 0.f;
#pragma unroll
    for (int w = 0; w < NWAVES; ++w) sv += red[w][tid];

    // Squash: coef = 1 - exp(-||S||); out = coef * S / (||S|| + eps)
    sq[tid] = sv * sv;
    __syncthreads();
    const int base = tid & ~15;                        // start of this b's 16 d-values
    float nrm2 = 0.f;
#pragma unroll
    for (int d = 0; d < DD; ++d) nrm2 += sq[base + d];
    const float nrm  = sqrtf(nrm2);
    const float coef = 1.0f - __expf(-nrm);
    const float o    = coef * sv / (nrm + 1e-7f);

    const int b = tid >> 4;
    const int d = tid & 15;
    out[((size_t)b * MCAPS + m) * DD + d] = o;
}

extern "C" void kernel_launch(void* const* d_in, const int* in_sizes, int n_in,
                              void* d_out, int out_size, void* d_ws, size_t ws_size,
                              hipStream_t stream) {
    const float* u  = (const float*)d_in[0];   // primary_caps [16,1152,8]
    const float* W  = (const float*)d_in[1];   // W            [10,1152,16,8]
    const float* Bp = (const float*)d_in[2];   // B_prior      [10,1,1152]
    float* out = (float*)d_out;                // [16,10,16]

    digitcaps_wmma_kernel<<<dim3(MCAPS), dim3(256), 0, stream>>>(u, W, Bp, out);
}